// convDU_7438883356881
// MI455X (gfx1250) — compile-verified
//
#include <hip/hip_runtime.h>
#include <stdint.h>

typedef float v2f __attribute__((ext_vector_type(2)));
typedef float v8f __attribute__((ext_vector_type(8)));
typedef int   v4i __attribute__((vector_size(16)));   // matches b128 builtin param type

#if defined(__gfx1250__) && __has_builtin(__builtin_amdgcn_global_load_async_to_lds_b32)
#define USE_ASYNC_LDS 1
#else
#define USE_ASYNC_LDS 0
#endif

#if USE_ASYNC_LDS && __has_builtin(__builtin_amdgcn_global_load_async_to_lds_b128)
#define HAVE_ASYNC_B128 1
#else
#define HAVE_ASYNC_B128 0
#endif

namespace {
constexpr int Cch = 2048;          // channels
constexpr int Hh  = 24;            // height (recurrence dim)
constexpr int Wd  = 8;             // width
constexpr int Nn  = 32;            // batch
constexpr int NJ  = Nn * Wd;       // 256 GEMM columns (j = n*8 + w)
constexpr int KHs = 9;             // conv kernel height
constexpr int CTR = 4;             // center tap
constexpr int CHW = Cch * Hh * Wd; // stride of n in NCHW
constexpr int HW  = Hh * Wd;       // stride of c in NCHW

constexpr int BLK_M  = 64;         // output rows per block
constexpr int BLK_J  = 64;         // output cols per block
constexpr int KC     = 64;         // K chunk staged in LDS
constexpr int KSLICE = 4;          // K split across blockIdx.z -> 512 blocks/step
constexpr int NTHR   = 128;        // 4 waves
constexpr int AS_S   = 80;         // A row stride: 2*80 % 64 == 32 -> lane halves bank-disjoint
constexpr int BS2_S  = 80;         // B float2 row stride (160 floats): same property
}

#if USE_ASYNC_LDS
__device__ __forceinline__ void async_cp4(const void* g, void* l) {
    __builtin_amdgcn_global_load_async_to_lds_b32(
        (__attribute__((address_space(1))) int*)(uintptr_t)g,
        (__attribute__((address_space(3))) int*)(uintptr_t)l, 0, 0);
}
#if HAVE_ASYNC_B128
__device__ __forceinline__ void async_cp16(const void* g, void* l) {
    __builtin_amdgcn_global_load_async_to_lds_b128(
        (__attribute__((address_space(1))) v4i*)(uintptr_t)g,
        (__attribute__((address_space(3))) v4i*)(uintptr_t)l, 0, 0);
}
#endif
__device__ __forceinline__ void async_wait0() {
#if __has_builtin(__builtin_amdgcn_s_wait_asynccnt)
    __builtin_amdgcn_s_wait_asynccnt(0);
#else
    asm volatile("s_wait_asynccnt 0" ::: "memory");
#endif
}
#endif

// ---- one-time weight pack + transpose: WpT[c][o] = W[o][c][4][0] ------------
// (A matrix rows are m=o, cols k=c; LDS wants [k][m] row-contiguous => b128)
__global__ void pack_wT_kernel(const float* __restrict__ W, float* __restrict__ WpT) {
    __shared__ float t[32][33];
    const int bo = blockIdx.x * 32;
    const int bc = blockIdx.y * 32;
    const int tx = threadIdx.x & 31, ty = threadIdx.x >> 5;  // 32x8
    #pragma unroll
    for (int yy = ty; yy < 32; yy += 8)
        t[yy][tx] = W[((bo + yy) * Cch + (bc + tx)) * KHs + CTR];
    __syncthreads();
    #pragma unroll
    for (int yy = ty; yy < 32; yy += 8)
        WpT[(bc + yy) * Cch + (bo + tx)] = t[tx][yy];        // coalesced over o
}

// ---- f0 = x[0]: copy h=0 slice of fea into out ------------------------------
__global__ void init_h0_kernel(const float* __restrict__ fea, float* __restrict__ out) {
    int idx = blockIdx.x * blockDim.x + threadIdx.x;   // 0 .. N*C*W-1
    int w = idx & 7;
    int c = (idx >> 3) & (Cch - 1);
    int n = idx >> 14;
    int addr = n * CHW + c * HW + /*h=0*/ 0 + w;       // identical layout in fea & out
    out[addr] = fea[addr];
}

// ---- K-sliced GEMM: partial[z][o][j] = Wc[o, kslice] @ stack[kslice, j] -----
__global__ __launch_bounds__(NTHR)
void step_gemm_kernel(const float* __restrict__ WpT,
                      const float* __restrict__ out,   // stack slice h_in (B matrix)
                      float* __restrict__ partial,
                      int h_in)
{
    __shared__ __align__(16) float As[KC][AS_S];       // [k][m], row-contiguous
    __shared__ __align__(16) v2f   Bs2[KC / 2][BS2_S]; // [k/2][j] = (B[k][j], B[k+1][j])

    const int tid  = threadIdx.x;
    const int lane = tid & 31;
    const int wave = tid >> 5;
    const int M0   = blockIdx.y * BLK_M;
    const int J0   = blockIdx.x * BLK_J;
    const int kb   = blockIdx.z * (Cch / KSLICE);
    const int lm   = lane & 15;
    const int hh   = lane >> 4;                // 0 = lanes 0-15, 1 = lanes 16-31
    const int kh2  = hh << 1;
    const int mb   = wave * 16;

    v8f acc[4] = {};

    for (int k0 = kb; k0 < kb + Cch / KSLICE; k0 += KC) {
        __syncthreads();
#if USE_ASYNC_LDS
#if HAVE_ASYNC_B128
        #pragma unroll
        for (int i = 0; i < (BLK_M * KC) / (4 * NTHR); ++i) {   // 8 x b128
            int q  = tid + i * NTHR;
            int k  = q >> 4;                 // 16 quads per 64-float row
            int qq = (q & 15) << 2;
            async_cp16(&WpT[(k0 + k) * Cch + M0 + qq], &As[k][qq]);
        }
#else
        #pragma unroll
        for (int i = 0; i < (BLK_M * KC) / NTHR; ++i) {
            int idx = tid + i * NTHR;
            int k = idx >> 6, m = idx & 63;
            async_cp4(&WpT[(k0 + k) * Cch + M0 + m], &As[k][m]);
        }
#endif
        #pragma unroll
        for (int i = 0; i < ((KC / 2) * BLK_J) / NTHR; ++i) {
            int idx = tid + i * NTHR;
            int p = idx >> 6, j = idx & (BLK_J - 1);
            int n = j >> 3, w = j & 7;
            const float* g = &out[n * CHW + (k0 + 2 * p) * HW + h_in * Wd + w];
            float* l = (float*)&Bs2[p][j];
            async_cp4(g,      l);
            async_cp4(g + HW, l + 1);
        }
        async_wait0();
#else
        #pragma unroll
        for (int i = 0; i < (BLK_M * KC) / NTHR; ++i) {
            int idx = tid + i * NTHR;
            int k = idx >> 6, m = idx & 63;
            As[k][m] = WpT[(k0 + k) * Cch + M0 + m];
        }
        #pragma unroll
        for (int i = 0; i < ((KC / 2) * BLK_J) / NTHR; ++i) {
            int idx = tid + i * NTHR;
            int p = idx >> 6, j = idx & (BLK_J - 1);
            int n = j >> 3, w = j & 7;
            const float* g = &out[n * CHW + (k0 + 2 * p) * HW + h_in * Wd + w];
            v2f t2; t2.x = g[0]; t2.y = g[HW];
            Bs2[p][j] = t2;
        }
#endif
        __syncthreads();

        #pragma unroll
        for (int kk = 0; kk < KC; kk += 4) {
            v2f a;
            a.x = As[kk + kh2][mb + lm];         // one ds_load_2addr_b32 (offset1:80)
            a.y = As[kk + kh2 + 1][mb + lm];
            const int p = (kk >> 1) + hh;
            #pragma unroll
            for (int t = 0; t < 4; ++t) {
                v2f b = Bs2[p][t * 16 + lm];     // b64 pairs, merged into 2addr_b64
                acc[t] = __builtin_amdgcn_wmma_f32_16x16x4_f32(
                    false, a, false, b, (short)0, acc[t], false, false);
            }
        }
    }

    // raw fp32 partial, layout [z][o][j]; D: VGPR r -> M=r / r+8, N=lane%16
    float* P = partial + (size_t)blockIdx.z * (Cch * NJ);
    const int oBase = M0 + mb + (hh << 3);
    #pragma unroll
    for (int r = 0; r < 8; ++r) {
        #pragma unroll
        for (int t = 0; t < 4; ++t)
            P[(oBase + r) * NJ + J0 + t * 16 + lm] = acc[t][r];
    }
}

// ---- deterministic K-slice reduce + bias + ReLU + residual ------------------
__global__ void step_reduce_kernel(const float* __restrict__ P,
                                   const float* __restrict__ bias,
                                   const float* __restrict__ fea,
                                   float* __restrict__ out,
                                   int h_out, int add_mode)
{
    int idx = blockIdx.x * blockDim.x + threadIdx.x;   // 0 .. C*NJ-1
    int o = idx >> 8, j = idx & 255;
    float s = P[idx];
    #pragma unroll
    for (int z = 1; z < KSLICE; ++z) s += P[idx + z * (Cch * NJ)];
    float v = fmaxf(s + bias[o], 0.0f);
    int n = j >> 3, w = j & 7;
    int addr = n * CHW + o * HW + h_out * Wd + w;
    float addv = (add_mode == 0) ? fea[addr] : out[addr];
    out[addr] = v + addv;
}

extern "C" void kernel_launch(void* const* d_in, const int* in_sizes, int n_in,
                              void* d_out, int out_size, void* d_ws, size_t ws_size,
                              hipStream_t stream) {
    const float* fea  = (const float*)d_in[0];   // (32, 2048, 24, 8)
    const float* Wf   = (const float*)d_in[1];   // (2048, 2048, 9, 1)
    const float* bias = (const float*)d_in[2];   // (2048,)
    float* out = (float*)d_out;                  // (32, 2048, 24, 8)
    float* WpT = (float*)d_ws;                   // 2048*2048 fp32 = 16 MB (transposed Wc)
    float* Prt = WpT + (size_t)Cch * Cch;        // KSLICE * 2048*256 fp32 = 8 MB partials

    (void)in_sizes; (void)n_in; (void)out_size; (void)ws_size;

    pack_wT_kernel<<<dim3(Cch / 32, Cch / 32), 256, 0, stream>>>(Wf, WpT);
    init_h0_kernel<<<(Nn * Cch * Wd) / 256, 256, 0, stream>>>(fea, out);

    const dim3 ggrid(NJ / BLK_J, Cch / BLK_M, KSLICE);  // (4, 32, 4) = 512 blocks
    const int  rgrid = (Cch * NJ) / 256;

    for (int h = 1; h < Hh; ++h) {               // forward scan
        step_gemm_kernel<<<ggrid, NTHR, 0, stream>>>(WpT, out, Prt, h - 1);
        step_reduce_kernel<<<rgrid, 256, 0, stream>>>(Prt, bias, fea, out, h, 0);
    }
    for (int h = Hh - 2; h >= 0; --h) {          // backward scan
        step_gemm_kernel<<<ggrid, NTHR, 0, stream>>>(WpT, out, Prt, h + 1);
        step_reduce_kernel<<<rgrid, 256, 0, stream>>>(Prt, bias, fea, out, h, 1);
    }
}